// RankNN_39402029973993
// MI455X (gfx1250) — compile-verified
//
#include <hip/hip_runtime.h>
#include <hip/hip_bf16.h>

typedef __attribute__((ext_vector_type(2))) float v2f;
typedef __attribute__((ext_vector_type(8))) float v8f;
typedef __attribute__((ext_vector_type(4))) unsigned int v4u;
typedef __attribute__((ext_vector_type(8))) int v8i;
typedef __attribute__((ext_vector_type(4))) int v4i;

#define L_    512
#define D_    768
#define KW    10
#define PDIM  50
#define RDIM  1586          // 2*D + P
#define NPAIR 10642         // L*(2K+1) - K*(K+1)
#define BATCH 4
#define MTOT  (BATCH * NPAIR)   // 42568
#define KPAD  1600              // RDIM rounded up to BK multiple
#define BM 128
#define BN 64
#define BK 16
#define NKITER (KPAD / BK)      // 100
#define LDA  20                 // LDS A row stride (floats), conflict padding
#define ABUF (BM * LDA)         // floats per A buffer
#define LDBP 84                 // LDS B row stride (floats): 64 data + 20 pad (TDM pad)
#define BBUF (BK * LDBP)        // floats per B buffer

#if __has_builtin(__builtin_amdgcn_tensor_load_to_lds) && __has_builtin(__builtin_amdgcn_s_wait_tensorcnt)
#define USE_TDM 1
#else
#define USE_TDM 0
#endif

// ---------------------------------------------------------------------------
// Build (ei, ci, rel) window-pair indices + emo_cau_pos output (1-based).
// ---------------------------------------------------------------------------
__global__ void idx_kernel(int* __restrict__ ei, int* __restrict__ ci,
                           int* __restrict__ rel, float* __restrict__ outpos) {
    int i = blockIdx.x;
    __shared__ int off;
    if (threadIdx.x == 0) {
        int o = 0;
        for (int t = 0; t < i; ++t)
            o += min(t + KW, L_ - 1) - max(t - KW, 0) + 1;
        off = o;
    }
    __syncthreads();
    int jlo = max(i - KW, 0), jhi = min(i + KW, L_ - 1);
    for (int idx = threadIdx.x; idx <= jhi - jlo; idx += blockDim.x) {
        int j = jlo + idx;
        int pos = off + idx;
        ei[pos] = i;
        ci[pos] = j;
        rel[pos] = j - i;
        outpos[2 * pos + 0] = (float)(i + 1);
        outpos[2 * pos + 1] = (float)(j + 1);
    }
}

// ---------------------------------------------------------------------------
// emb = kernel @ pos_emb collapses to 21 distinct rows (rel multiplicity L-|r|)
// ---------------------------------------------------------------------------
__global__ void emb_kernel(const float* __restrict__ pos_emb,
                           float* __restrict__ emb21) {
    for (int idx = threadIdx.x; idx < 21 * PDIM; idx += blockDim.x) {
        int ri = idx / PDIM;
        int p  = idx - ri * PDIM;
        float s = 0.f;
        for (int rq = 0; rq < 21; ++rq) {
            float d = (float)(ri - rq);
            float w = expf(-d * d) * (float)(L_ - abs(rq - KW));
            s += w * pos_emb[rq * PDIM + p];
        }
        emb21[idx] = s;
    }
}

#if USE_TDM
// Issue a TDM load of the 16x64 W1 tile starting at (k0, N0) into LDS.
// tensor_dim0/1 = remaining extent from tile origin -> HW zero-fills OOB.
// pad: every 64 DWORDs insert 20 DWORDs -> LDS row stride = 84 floats.
__device__ __forceinline__ void tdm_load_B(const float* W1, int N0, int k0,
                                           unsigned lds_addr) {
    unsigned long long ga =
        (unsigned long long)(uintptr_t)(W1 + (size_t)k0 * RDIM + N0);
    int dim0 = RDIM - N0;   // cols remaining
    int dim1 = RDIM - k0;   // rows remaining
    v4u g0;
    g0[0] = 1u;                                   // count=1, no gather
    g0[1] = lds_addr;                             // LDS byte address
    g0[2] = (unsigned)ga;                         // global_addr[31:0]
    g0[3] = (unsigned)((ga >> 32) & 0x01FFFFFFu)  // global_addr[56:32]
            | 0x80000000u;                        // type=2 ("image")
    v8i g1;
    g1[0] = 0x27520000;                 // data_size=4B | pad_en | intv=5 | amt=19(+1)
    g1[1] = (dim0 & 0xFFFF) << 16;      // tensor_dim0[15:0] (hi bits zero)
    g1[2] = (dim1 & 0xFFFF) << 16;      // dim0[31:16]=0 | tensor_dim1[15:0]
    g1[3] = BN << 16;                   // tensor_dim1 hi=0 | tile_dim0=64
    g1[4] = BK;                         // tile_dim1=16, tile_dim2=0
    g1[5] = RDIM;                       // tensor_dim0_stride lo
    g1[6] = 0;                          // stride hi / dim1_stride lo
    g1[7] = 0;
    v4i z4 = {0, 0, 0, 0};
    v8i z8 = {0, 0, 0, 0, 0, 0, 0, 0};
    __builtin_amdgcn_tensor_load_to_lds(g0, g1, z4, z4, z8, 0);
}
#endif

// ---------------------------------------------------------------------------
// Fused GEMM: h = relu(couples @ W1 + b1), couples gathered on the fly.
// 128x64x16 tiles, 8 wave32 (4x2), 2x2 16x16 f32 WMMA tiles per wave.
// A and B both double-buffered in LDS -> single barrier per K step.
// B tiles move via the Tensor Data Mover (async DMA overlaps the WMMAs).
// ---------------------------------------------------------------------------
__global__ __launch_bounds__(256)
void rank_gemm(const float* __restrict__ doc, const float* __restrict__ W1,
               const float* __restrict__ b1, const float* __restrict__ emb21,
               const int* __restrict__ ei, const int* __restrict__ ci,
               const int* __restrict__ rel, float* __restrict__ hout) {
    __shared__ __align__(16) float As[2 * ABUF];
    __shared__ __align__(16) float Bs[2 * BBUF];
    __shared__ int baseE[BM], baseC[BM], baseR[BM];

    const int t  = threadIdx.x;
    const int M0 = blockIdx.y * BM;
    const int N0 = blockIdx.x * BN;

    // Rows >= MTOT are clamped: they load valid (unused) data; the epilogue
    // masks their stores. This keeps the A gather free of per-lane EXEC ops.
    if (t < BM) {
        int m = min(M0 + t, MTOT - 1);
        int b = m / NPAIR;
        int p = m - b * NPAIR;
        baseE[t] = (b * L_ + ei[p]) * D_;
        baseC[t] = (b * L_ + ci[p]) * D_;
        baseR[t] = (rel[p] + KW) * PDIM;
    }
    __syncthreads();

    // A gather: one float4 of K-columns of virtual row couples[m][k..k+3].
    // The segment branch is wave-uniform (16-wide K chunks never straddle).
    auto loadA4 = [&](int slot, int k0) -> float4 {
        int row = slot >> 2;
        int k   = k0 + ((slot & 3) << 2);
        float4 v;
        if (k < D_) {
            v = *(const float4*)(doc + baseE[row] + k);
        } else if (k < 2 * D_) {
            v = *(const float4*)(doc + baseC[row] + (k - D_));
        } else {
            int kk = k - 2 * D_;
            int bR = baseR[row];
            v.x = (kk + 0 < PDIM) ? emb21[bR + kk + 0] : 0.f;
            v.y = (kk + 1 < PDIM) ? emb21[bR + kk + 1] : 0.f;
            v.z = (kk + 2 < PDIM) ? emb21[bR + kk + 2] : 0.f;
            v.w = (kk + 3 < PDIM) ? emb21[bR + kk + 3] : 0.f;
        }
        return v;
    };
    auto storeA = [&](float* Ab, float4 x0, float4 x1) {
        int row = t >> 2, kq = t & 3;
        *(float4*)&Ab[row * LDA + (kq << 2)] = x0;
        int s2 = t + 256;
        row = s2 >> 2; kq = s2 & 3;
        *(float4*)&Ab[row * LDA + (kq << 2)] = x1;
    };

    const int lane = t & 31;
    const int wid  = t >> 5;
    const int wM   = (wid & 3) << 5;   // 0,32,64,96
    const int wN   = (wid >> 2) << 5;  // 0,32
    const int half = lane >> 4;
    const int l15  = lane & 15;

#if USE_TDM
    const unsigned bsbase = (unsigned)(uintptr_t)&Bs[0];  // LDS byte offset
    if (wid == 0) tdm_load_B(W1, N0, 0, bsbase);
#endif
    // prologue: stage K-chunk 0 into buffer 0
    {
        float4 ra0 = loadA4(t, 0), ra1 = loadA4(t + 256, 0);
        storeA(&As[0], ra0, ra1);
    }
#if USE_TDM
    if (wid == 0) __builtin_amdgcn_s_wait_tensorcnt(0);
#else
    {
        float* Bb = &Bs[0];
        int kk = t >> 4, c = (t & 15) << 2;
        const float* wp = W1 + (size_t)kk * RDIM + N0 + c;
        *(float2*)&Bb[kk * LDBP + c]     = *(const float2*)(wp);
        *(float2*)&Bb[kk * LDBP + c + 2] = *(const float2*)(wp + 2);
    }
#endif
    __syncthreads();

    v8f acc00 = {}, acc01 = {}, acc10 = {}, acc11 = {};

    for (int kt = 0; kt < NKITER; ++kt) {
        const int cur = kt & 1, nxt = cur ^ 1;
        float4 ra0, ra1;
#if !USE_TDM
        float2 rb0 = make_float2(0.f, 0.f), rb1 = make_float2(0.f, 0.f);
#endif
        if (kt + 1 < NKITER) {
            int k0 = (kt + 1) * BK;
            ra0 = loadA4(t, k0);            // global prefetch, overlaps WMMAs
            ra1 = loadA4(t + 256, k0);
#if USE_TDM
            if (wid == 0)                   // async DMA into the other buffer
                tdm_load_B(W1, N0, k0, bsbase + (unsigned)(nxt * BBUF * 4));
#else
            int kk = t >> 4, c = (t & 15) << 2;
            int krow = k0 + kk, col = N0 + c;
            if (krow < RDIM) {
                const float* wp = W1 + (size_t)krow * RDIM + col;
                if (col + 2 <= RDIM) rb0 = *(const float2*)(wp);
                if (col + 4 <= RDIM) rb1 = *(const float2*)(wp + 2);
            }
#endif
        }

        const float* Ac = &As[cur * ABUF];
        const float* Bc = &Bs[cur * BBUF];
#pragma unroll
        for (int ks = 0; ks < 4; ++ks) {
            int ko = (ks << 2) + (half << 1);
            const float* ap0 = &Ac[(wM + l15) * LDA + ko];
            const float* ap1 = &Ac[(wM + 16 + l15) * LDA + ko];
            v2f a0, a1, b0, b1f;
            a0[0] = ap0[0];                 a0[1] = ap0[1];
            a1[0] = ap1[0];                 a1[1] = ap1[1];
            b0[0]  = Bc[ko * LDBP + wN + l15];
            b0[1]  = Bc[(ko + 1) * LDBP + wN + l15];
            b1f[0] = Bc[ko * LDBP + wN + 16 + l15];
            b1f[1] = Bc[(ko + 1) * LDBP + wN + 16 + l15];
            acc00 = __builtin_amdgcn_wmma_f32_16x16x4_f32(false, a0, false, b0,  (short)0, acc00, false, false);
            acc01 = __builtin_amdgcn_wmma_f32_16x16x4_f32(false, a0, false, b1f, (short)0, acc01, false, false);
            acc10 = __builtin_amdgcn_wmma_f32_16x16x4_f32(false, a1, false, b0,  (short)0, acc10, false, false);
            acc11 = __builtin_amdgcn_wmma_f32_16x16x4_f32(false, a1, false, b1f, (short)0, acc11, false, false);
        }

        if (kt + 1 < NKITER) {
            storeA(&As[nxt * ABUF], ra0, ra1);   // commit next A tile
#if !USE_TDM
            float* Bb = &Bs[nxt * BBUF];
            int kk = t >> 4, c = (t & 15) << 2;
            *(float2*)&Bb[kk * LDBP + c]     = rb0;
            *(float2*)&Bb[kk * LDBP + c + 2] = rb1;
#else
            if (wid == 0) __builtin_amdgcn_s_wait_tensorcnt(0);
#endif
        }
        __syncthreads();   // single barrier per K step
    }

    // epilogue: bias + relu, masked store of h (C layout: M = vr + 8*half)
    int colA = N0 + wN + l15;
    int colB = colA + 16;
    float b1a = (colA < RDIM) ? b1[colA] : 0.f;
    float b1b = (colB < RDIM) ? b1[colB] : 0.f;
#pragma unroll
    for (int vr = 0; vr < 8; ++vr) {
        int mA = M0 + wM + vr + (half << 3);
        int mB = mA + 16;
        if (mA < MTOT) {
            if (colA < RDIM) hout[(size_t)mA * RDIM + colA] = fmaxf(acc00[vr] + b1a, 0.f);
            if (colB < RDIM) hout[(size_t)mA * RDIM + colB] = fmaxf(acc01[vr] + b1b, 0.f);
        }
        if (mB < MTOT) {
            if (colA < RDIM) hout[(size_t)mB * RDIM + colA] = fmaxf(acc10[vr] + b1a, 0.f);
            if (colB < RDIM) hout[(size_t)mB * RDIM + colB] = fmaxf(acc11[vr] + b1b, 0.f);
        }
    }
}

// ---------------------------------------------------------------------------
// pred = h @ W2 + b2 : one wave32 per output row, deterministic reduction.
// ---------------------------------------------------------------------------
__global__ __launch_bounds__(256)
void pred_kernel(const float* __restrict__ h, const float* __restrict__ W2,
                 const float* __restrict__ b2, float* __restrict__ pred) {
    int gw   = (int)((blockIdx.x * blockDim.x + threadIdx.x) >> 5);
    int lane = threadIdx.x & 31;
    if (gw >= MTOT) return;
    const float* hr = h + (size_t)gw * RDIM;
    float s = 0.f;
    for (int r = lane; r < RDIM; r += 32) s += hr[r] * W2[r];
#pragma unroll
    for (int m = 16; m >= 1; m >>= 1) s += __shfl_xor(s, m, 32);
    if (lane == 0) pred[gw] = s + b2[0];
}

extern "C" void kernel_launch(void* const* d_in, const int* in_sizes, int n_in,
                              void* d_out, int out_size, void* d_ws, size_t ws_size,
                              hipStream_t stream) {
    (void)in_sizes; (void)n_in; (void)out_size; (void)ws_size;
    const float* doc  = (const float*)d_in[0];   // [4,512,768]
    const float* pose = (const float*)d_in[1];   // [21,50]
    const float* W1   = (const float*)d_in[2];   // [1586,1586]
    const float* b1   = (const float*)d_in[3];   // [1586]
    const float* W2   = (const float*)d_in[4];   // [1586,1]
    const float* b2   = (const float*)d_in[5];   // [1]

    float* out    = (float*)d_out;
    float* pred   = out;                          // [4*10642]
    float* outpos = out + MTOT;                   // [10642*2] (as floats)
    float* hout   = outpos + 2 * NPAIR;           // [4*10642*1586]

    int*   ei    = (int*)d_ws;
    int*   ci    = ei + NPAIR;
    int*   rel   = ci + NPAIR;
    float* emb21 = (float*)(rel + NPAIR);         // [21*50]

    idx_kernel<<<L_, 32, 0, stream>>>(ei, ci, rel, outpos);
    emb_kernel<<<1, 256, 0, stream>>>(pose, emb21);

    dim3 grid((RDIM + BN - 1) / BN, (MTOT + BM - 1) / BM);  // 25 x 333
    rank_gemm<<<grid, 256, 0, stream>>>(doc, W1, b1, emb21, ei, ci, rel, hout);

    pred_kernel<<<(MTOT * 32 + 255) / 256, 256, 0, stream>>>(hout, W2, b2, pred);
}